// LogicityPredictorVis_61057255080038
// MI455X (gfx1250) — compile-verified
//
#include <hip/hip_runtime.h>
#include <hip/hip_bf16.h>

typedef float v2f __attribute__((ext_vector_type(2)));
typedef float v8f __attribute__((ext_vector_type(8)));

// ---------------------------------------------------------------------------
// f32 WMMA: D[16x16] = A[16x4] * B[4x16] + C   (V_WMMA_F32_16X16X4_F32)
//   A: lanes 0-15 -> rows M=lane, vgpr{0,1} = K{k,k+1}; lanes 16-31 -> K{k+2,k+3}
//   B: lanes 0-15 -> cols N=lane, vgpr{0,1} = K{k,k+1}; lanes 16-31 -> K{k+2,k+3}
//   C/D: vgpr r: lanes 0-15 hold row r, lanes 16-31 hold row r+8
// Only D rows 0-7 are valid/stored, so A rows 8-15 may hold anything: we map
// lane row -> (lh & 7), eliminating all predication inside the WMMA loop.
// ---------------------------------------------------------------------------
static __device__ __forceinline__ v8f wmma4(v2f a, v2f b, v8f c) {
  return __builtin_amdgcn_wmma_f32_16x16x4_f32(
      /*neg_a=*/false, a, /*neg_b=*/false, b,
      /*c_mod=*/(short)0, c, /*reuse_a=*/false, /*reuse_b=*/false);
}

// Hout[8][Nout] = act(Ain[8][Kin] @ W[Kin][Nout] + bvec),  act 0=relu, 1=sigmoid
// Ain rows are 8-aligned in memory; v2f loads require 8B alignment (Kin even).
static __device__ __forceinline__ void mlp_layer_wmma(
    const float* __restrict__ Ain, const float* __restrict__ W,
    const float* __restrict__ bvec, float* __restrict__ Hout,
    int Kin, int Nout, int act, int tid, int nthreads) {
  const int lane   = tid & 31;
  const int wave   = tid >> 5;
  const int nwaves = nthreads >> 5;
  const int lh     = lane & 15;
  const bool hi    = lane >= 16;
  const int koff   = hi ? 2 : 0;
  const int ntiles = Nout >> 4;
  const float* arow = Ain + (lh & 7) * Kin + koff;  // rows 8-15 mirror 0-7

  for (int nt = wave; nt < ntiles; nt += nwaves) {
    const int n0 = nt << 4;
    const float* wcol = W + n0 + lh;
    __builtin_prefetch(wcol, 0, 1);
    v8f acc = {0.f, 0.f, 0.f, 0.f, 0.f, 0.f, 0.f, 0.f};

    // software pipeline: fetch k+4 fragments while wmma(k) issues
    v2f a0 = *(const v2f*)(arow);
    v2f b0;
    b0.x = wcol[(size_t)koff * Nout];
    b0.y = wcol[(size_t)(koff + 1) * Nout];
#pragma unroll 4
    for (int k = 0; k < Kin - 4; k += 4) {
      const int kn = k + 4;
      v2f a1 = *(const v2f*)(arow + kn);
      v2f b1;
      b1.x = wcol[(size_t)(kn + koff) * Nout];
      b1.y = wcol[(size_t)(kn + koff + 1) * Nout];
      acc = wmma4(a0, b0, acc);
      a0 = a1;
      b0 = b1;
    }
    acc = wmma4(a0, b0, acc);

    if (!hi) {                           // low lanes hold the 8 valid rows
      const float bb = bvec[n0 + lh];
#pragma unroll
      for (int r = 0; r < 8; ++r) {
        float v = acc[r] + bb;
        v = (act == 0) ? fmaxf(v, 0.f) : 1.f / (1.f + __expf(-v));
        Hout[r * Nout + n0 + lh] = v;
      }
    }
  }
}

// ---------------------------------------------------------------------------
// Kernel 1: node-concept MLP on batch element 0 only.
// nc[8][2048] = sigmoid(relu(relu(roi0 @ W1 + b1) @ W2 + b2) @ W3 + b3)
// ---------------------------------------------------------------------------
__global__ void __launch_bounds__(256) node_mlp_kernel(
    const float* __restrict__ roi,
    const float* __restrict__ w1, const float* __restrict__ b1,
    const float* __restrict__ w2, const float* __restrict__ b2,
    const float* __restrict__ w3, const float* __restrict__ b3,
    float* __restrict__ nc) {
  __shared__ __align__(16) float sA[8 * 1024];   // 32 KB
  __shared__ __align__(16) float sH1[8 * 512];   // 16 KB
  __shared__ __align__(16) float sH2[8 * 256];   //  8 KB
  const int tid = threadIdx.x;
  for (int i = tid; i < 8 * 1024; i += 256) sA[i] = roi[i];  // batch 0 slice
  __syncthreads();
  mlp_layer_wmma(sA,  w1, b1, sH1, 1024, 512,  0, tid, 256);
  __syncthreads();
  mlp_layer_wmma(sH1, w2, b2, sH2, 512,  256,  0, tid, 256);
  __syncthreads();
  mlp_layer_wmma(sH2, w3, b3, nc,  256,  2048, 1, tid, 256);
}

// ---------------------------------------------------------------------------
// Kernel 2: edge-attribute path (batch 0) + small contractions.
//   t_pad[64][128] = relu(e4 @ pr_w1 + pr_b1)   (rows 56..63 zero)
//   qb[8][4] = X @ pr_b2(view [2048,4]);  xr[8][4] = X @ root
// ---------------------------------------------------------------------------
__global__ void __launch_bounds__(64) edge_prep_kernel(
    const float* __restrict__ bbox, const float* __restrict__ dirv,
    const float* __restrict__ pri,
    const float* __restrict__ ew1, const float* __restrict__ eb1,
    const float* __restrict__ ew2, const float* __restrict__ eb2,
    const float* __restrict__ ew3, const float* __restrict__ eb3,
    const float* __restrict__ pw1, const float* __restrict__ pb1,
    const float* __restrict__ pb2, const float* __restrict__ root,
    const float* __restrict__ nc,
    float* __restrict__ t_pad, float* __restrict__ qb, float* __restrict__ xr) {
  __shared__ float s_attr[8 * 8];
  __shared__ float s_g1[8 * 256];
  __shared__ float s_g2[8 * 64];
  __shared__ float s_ea[8 * 21];
  __shared__ float s_e4[56 * 4];
  const int tid = threadIdx.x;

  {  // node_attr = [bbox/1024, directions]
    int n = tid >> 3, q = tid & 7;
    s_attr[n * 8 + q] = (q < 4) ? bbox[n * 4 + q] * (1.f / 1024.f)
                                : dirv[n * 4 + (q - 4)];
  }
  __syncthreads();
  for (int i = tid; i < 8 * 256; i += 64) {          // g1 = relu(attr@ew1+eb1)
    int n = i >> 8, c = i & 255;
    float a = eb1[c];
    for (int q = 0; q < 8; ++q) a += s_attr[n * 8 + q] * ew1[q * 256 + c];
    s_g1[i] = fmaxf(a, 0.f);
  }
  __syncthreads();
  for (int i = tid; i < 8 * 64; i += 64) {           // g2 = relu(g1@ew2+eb2)
    int n = i >> 6, c = i & 63;
    float a = eb2[c];
    for (int q = 0; q < 256; ++q) a += s_g1[n * 256 + q] * ew2[q * 64 + c];
    s_g2[i] = fmaxf(a, 0.f);
  }
  __syncthreads();
  for (int i = tid; i < 8 * 21; i += 64) {           // ea = sigmoid(g2@ew3+eb3)
    int n = i / 21, c = i % 21;
    float a = eb3[c];
    for (int q = 0; q < 64; ++q) a += s_g2[n * 64 + q] * ew3[q * 21 + c];
    s_ea[i] = 1.f / (1.f + __expf(-a));
  }
  __syncthreads();
  if (tid < 56) {  // e4[e] = {ea[i][3c..3c+2], (p[i] > p[c])}, e = i*7 + c
    int i = tid / 7, c = tid % 7;
    s_e4[tid * 4 + 0] = s_ea[i * 21 + 3 * c + 0];
    s_e4[tid * 4 + 1] = s_ea[i * 21 + 3 * c + 1];
    s_e4[tid * 4 + 2] = s_ea[i * 21 + 3 * c + 2];
    s_e4[tid * 4 + 3] = (pri[i] > pri[c]) ? 1.f : 0.f;
  }
  __syncthreads();
  for (int i = tid; i < 64 * 128; i += 64) {         // t = relu(e4@pw1+pb1)
    int e = i >> 7, k = i & 127;
    float v = 0.f;
    if (e < 56) {
      v = pb1[k];
      for (int q = 0; q < 4; ++q) v += s_e4[e * 4 + q] * pw1[q * 128 + k];
      v = fmaxf(v, 0.f);
    }
    t_pad[i] = v;
  }
  if (tid < 32) {                                    // qb, xr
    int s = tid >> 2, o = tid & 3;
    float aq = 0.f, ar = 0.f;
    for (int c = 0; c < 2048; ++c) {
      float xv = nc[s * 2048 + c];
      aq += xv * pb2[c * 4 + o];
      ar += xv * root[c * 4 + o];
    }
    qb[tid] = aq;
    xr[tid] = ar;
  }
}

// ---------------------------------------------------------------------------
// Kernel 3: Q[s, k*4+o] = sum_c X[s,c] * pr_w2[k, c*4+o]
// Q2[16x512] = Xmir[16x2048] @ P[2048x512], P[c,j] = pw2[(j>>2)*8192 + 4c + (j&3)]
// 4 blocks x 8 waves, one 16x16 tile per wave, K = 2048 (512 wmma steps).
// ---------------------------------------------------------------------------
__global__ void __launch_bounds__(256) qgemm_kernel(
    const float* __restrict__ nc, const float* __restrict__ pw2,
    float* __restrict__ q2) {
  const int tid  = threadIdx.x;
  const int lane = tid & 31;
  const int wave = tid >> 5;
  const int lh   = lane & 15;
  const bool hi  = lane >= 16;
  const int koff = hi ? 2 : 0;
  const float* arow = nc + (lh & 7) * 2048 + koff;   // rows 8-15 mirror 0-7

  for (int nt = blockIdx.x * 8 + wave; nt < 32; nt += gridDim.x * 8) {
    const int n0 = nt << 4;
    const int j  = n0 + lh;                       // output column for this lane
    const float* bcol = pw2 + (size_t)(j >> 2) * 8192 + (j & 3);  // elem c at bcol[4c]
    __builtin_prefetch(bcol, 0, 1);
    v8f acc = {0.f, 0.f, 0.f, 0.f, 0.f, 0.f, 0.f, 0.f};

    v2f a0 = *(const v2f*)(arow);
    v2f b0;
    b0.x = bcol[(size_t)koff * 4];
    b0.y = bcol[(size_t)(koff + 1) * 4];
#pragma unroll 4
    for (int k = 0; k < 2048 - 4; k += 4) {
      const int kn = k + 4;
      v2f a1 = *(const v2f*)(arow + kn);
      v2f b1;
      b1.x = bcol[(size_t)(kn + koff) * 4];
      b1.y = bcol[(size_t)(kn + koff + 1) * 4];
      acc = wmma4(a0, b0, acc);
      a0 = a1;
      b0 = b1;
    }
    acc = wmma4(a0, b0, acc);

    if (!hi) {
#pragma unroll
      for (int r = 0; r < 8; ++r) q2[r * 512 + j] = acc[r];
    }
  }
}

// ---------------------------------------------------------------------------
// Kernel 4: msg[e,o] = qb[src,o] + sum_k t[e,k]*Q[src,k*4+o];
// segment-sum over DST via LDS float atomics; out = agg + x@root + bias.
// ---------------------------------------------------------------------------
__global__ void __launch_bounds__(64) finale_kernel(
    const float* __restrict__ t_pad, const float* __restrict__ q2,
    const float* __restrict__ qb, const float* __restrict__ xr,
    const float* __restrict__ bias, float* __restrict__ out) {
  __shared__ float s_agg[32];
  const int tid = threadIdx.x;
  if (tid < 32) s_agg[tid] = 0.f;
  __syncthreads();
  if (tid < 56) {
    const int e = tid;
    const int i = e / 7, idx = e % 7;
    const int j = (idx < i) ? idx : idx + 1;      // DST (skip self-loop)
    const int src = i;                            // SRC
    float m0 = qb[src * 4 + 0], m1 = qb[src * 4 + 1];
    float m2 = qb[src * 4 + 2], m3 = qb[src * 4 + 3];
    for (int k = 0; k < 128; ++k) {
      const float tv = t_pad[e * 128 + k];
      m0 += tv * q2[src * 512 + k * 4 + 0];
      m1 += tv * q2[src * 512 + k * 4 + 1];
      m2 += tv * q2[src * 512 + k * 4 + 2];
      m3 += tv * q2[src * 512 + k * 4 + 3];
    }
    atomicAdd(&s_agg[j * 4 + 0], m0);
    atomicAdd(&s_agg[j * 4 + 1], m1);
    atomicAdd(&s_agg[j * 4 + 2], m2);
    atomicAdd(&s_agg[j * 4 + 3], m3);
  }
  __syncthreads();
  if (tid < 32) out[tid] = s_agg[tid] + xr[tid] + bias[tid & 3];
}

// ---------------------------------------------------------------------------
extern "C" void kernel_launch(void* const* d_in, const int* in_sizes, int n_in,
                              void* d_out, int out_size, void* d_ws, size_t ws_size,
                              hipStream_t stream) {
  (void)in_sizes; (void)n_in; (void)out_size; (void)ws_size;
  const float* roi  = (const float*)d_in[0];
  const float* bbox = (const float*)d_in[1];
  const float* dirv = (const float*)d_in[2];
  const float* pri  = (const float*)d_in[3];
  const float* ncw1 = (const float*)d_in[4];
  const float* ncb1 = (const float*)d_in[5];
  const float* ncw2 = (const float*)d_in[6];
  const float* ncb2 = (const float*)d_in[7];
  const float* ncw3 = (const float*)d_in[8];
  const float* ncb3 = (const float*)d_in[9];
  const float* epw1 = (const float*)d_in[10];
  const float* epb1 = (const float*)d_in[11];
  const float* epw2 = (const float*)d_in[12];
  const float* epb2 = (const float*)d_in[13];
  const float* epw3 = (const float*)d_in[14];
  const float* epb3 = (const float*)d_in[15];
  const float* prw1 = (const float*)d_in[16];
  const float* prb1 = (const float*)d_in[17];
  const float* prw2 = (const float*)d_in[18];
  const float* prb2 = (const float*)d_in[19];
  const float* root = (const float*)d_in[20];
  const float* bias = (const float*)d_in[21];

  float* ws   = (float*)d_ws;
  float* nc   = ws;              // [8][2048]  = 16384 floats
  float* tpad = ws + 16384;      // [64][128]  =  8192 floats
  float* q2   = ws + 24576;      // [8][512]   =  4096 floats
  float* qb   = ws + 28672;      // [8][4]
  float* xr   = ws + 28704;      // [8][4]

  node_mlp_kernel<<<1, 256, 0, stream>>>(roi, ncw1, ncb1, ncw2, ncb2, ncw3, ncb3, nc);
  edge_prep_kernel<<<1, 64, 0, stream>>>(bbox, dirv, pri, epw1, epb1, epw2, epb2,
                                         epw3, epb3, prw1, prb1, prb2, root, nc,
                                         tpad, qb, xr);
  qgemm_kernel<<<4, 256, 0, stream>>>(nc, prw2, q2);
  finale_kernel<<<1, 64, 0, stream>>>(tpad, q2, qb, xr, bias, (float*)d_out);
}